// GaussianBlur2D_87127706567237
// MI455X (gfx1250) — compile-verified
//
#include <hip/hip_runtime.h>

// Separable 101-tap Gaussian blur on 4096x4096 fp32.
// Each 1D pass = Toeplitz-band matmuls on V_WMMA_F32_16X16X4_F32 (fp32-exact).
// Each wave computes a 16x64 strip (4 tiles along the conv axis): one streamed
// 164-element window (41 chunks) feeds 4 shifted Toeplitz accumulator chains,
// cutting load amplification from 7.25x to 2.56x and giving 4-way WMMA ILP.

#define IMG     4096
#define KS      101
#define PAD     50
#define NCHUNK  29                         // K = 116 per tile, chunks of 4
#define TBLK    4                          // tiles per wave along conv axis
#define NSTREAM (NCHUNK + 4 * (TBLK - 1))  // 41 chunks = 164 elements

typedef __attribute__((ext_vector_type(2))) float v2f;
typedef __attribute__((ext_vector_type(8))) float v8f;

// Branch-free Toeplitz entry: clamped LDS read + select-to-zero.
__device__ __forceinline__ float toep(const float* gs, int idx) {
    bool ok = (unsigned)idx < (unsigned)KS;
    float v = gs[ok ? idx : 0];
    return ok ? v : 0.0f;
}

// ---------------- horizontal pass: tmp[r,c] = sum_k g[k] * x[r, c+k-50] ----
// A = streamed image rows, B = constant Toeplitz g. Strip: rows r0..r0+15,
// cols q0..q0+63 (4 tiles).
template <bool GUARD>
__device__ __forceinline__ void hblur_strip(const float* __restrict__ x,
                                            float* __restrict__ tmp,
                                            const float* gs,
                                            int r0, int q0, int lane) {
    const int m16  = lane & 15;
    const int koff = (lane >> 4) << 1;      // 0 (lanes 0-15) or 2 (lanes 16-31)

    // Toeplitz B fragments: B[m, j] = g[m - j], j = m16
    v2f bfrag[NCHUNK];
#pragma unroll
    for (int kk = 0; kk < NCHUNK; ++kk) {
        int m = kk * 4 + koff - m16;
        bfrag[kk].x = toep(gs, m);
        bfrag[kk].y = toep(gs, m + 1);
    }

    const float* rowp  = x + (size_t)(r0 + m16) * IMG;
    const int    cbase = q0 - PAD + koff;   // even -> 8-byte aligned b64 loads

    v8f acc[TBLK] = {};
#pragma unroll
    for (int m = 0; m < NSTREAM; ++m) {
        v2f s;
        if (!GUARD) {
            s = *(const v2f*)(rowp + cbase + 4 * m);
        } else {
            int  cc  = cbase + 4 * m;
            bool ok0 = (unsigned)cc       < (unsigned)IMG;
            bool ok1 = (unsigned)(cc + 1) < (unsigned)IMG;
            float v0 = rowp[ok0 ? cc     : 0];
            float v1 = rowp[ok1 ? cc + 1 : 0];
            s.x = ok0 ? v0 : 0.0f;
            s.y = ok1 ? v1 : 0.0f;
        }
#pragma unroll
        for (int t = 0; t < TBLK; ++t) {
            int rel = m - 4 * t;            // tile t uses chunks [4t, 4t+28]
            if (rel >= 0 && rel < NCHUNK)
                acc[t] = __builtin_amdgcn_wmma_f32_16x16x4_f32(
                    false, s, false, bfrag[rel], (short)0, acc[t], false, false);
        }
    }

#pragma unroll
    for (int t = 0; t < TBLK; ++t) {
        float* op = tmp + (size_t)(r0 + ((lane >> 4) << 3)) * IMG
                        + (q0 + 16 * t + m16);
#pragma unroll
        for (int v = 0; v < 8; ++v)
            op[v * IMG] = acc[t][v];
    }
}

// ---------------- vertical pass: out[r,c] = sum_k g[k] * tmp[r+k-50, c] ----
// A = constant Toeplitz g, B = streamed column patch. Strip: rows q0..q0+63
// (4 tiles), cols c0..c0+15.
template <bool GUARD>
__device__ __forceinline__ void vblur_strip(const float* __restrict__ tmp,
                                            float* __restrict__ out,
                                            const float* gs,
                                            int q0, int c0, int lane) {
    const int m16  = lane & 15;
    const int koff = (lane >> 4) << 1;

    // Toeplitz A fragments: A[i, m] = g[m - i], i = m16
    v2f afrag[NCHUNK];
#pragma unroll
    for (int kk = 0; kk < NCHUNK; ++kk) {
        int m = kk * 4 + koff - m16;
        afrag[kk].x = toep(gs, m);
        afrag[kk].y = toep(gs, m + 1);
    }

    const float* colp  = tmp + (c0 + m16);
    const int    rbase = q0 - PAD + koff;

    v8f acc[TBLK] = {};
    const float* bp = colp + (size_t)(GUARD ? 0 : rbase) * IMG;
#pragma unroll
    for (int m = 0; m < NSTREAM; ++m) {
        v2f s;
        if (!GUARD) {
            // row offsets up to 163*16KB = 2.6MB: fit the signed 24-bit imm
            s.x = bp[(4 * m)     * IMG];
            s.y = bp[(4 * m + 1) * IMG];
        } else {
            int  rr  = rbase + 4 * m;
            bool ok0 = (unsigned)rr       < (unsigned)IMG;
            bool ok1 = (unsigned)(rr + 1) < (unsigned)IMG;
            float v0 = colp[(size_t)(ok0 ? rr     : 0) * IMG];
            float v1 = colp[(size_t)(ok1 ? rr + 1 : 0) * IMG];
            s.x = ok0 ? v0 : 0.0f;
            s.y = ok1 ? v1 : 0.0f;
        }
#pragma unroll
        for (int t = 0; t < TBLK; ++t) {
            int rel = m - 4 * t;
            if (rel >= 0 && rel < NCHUNK)
                acc[t] = __builtin_amdgcn_wmma_f32_16x16x4_f32(
                    false, afrag[rel], false, s, (short)0, acc[t], false, false);
        }
    }

#pragma unroll
    for (int t = 0; t < TBLK; ++t) {
        float* op = out + (size_t)(q0 + 16 * t + ((lane >> 4) << 3)) * IMG
                        + (c0 + m16);
#pragma unroll
        for (int v = 0; v < 8; ++v)
            op[v * IMG] = acc[t][v];
    }
}

// g[k] = w[k,50] * rsqrt(w[50,50])  (exact factorization of outer(g,g))
__device__ __forceinline__ void load_g(const float* __restrict__ w, float* gs, int t) {
    if (t < KS) {
        float inv = __frsqrt_rn(w[PAD * KS + PAD]);
        gs[t] = w[t * KS + PAD] * inv;
    }
    __syncthreads();
}

__global__ void __launch_bounds__(256)
gauss_hpass(const float* __restrict__ x, const float* __restrict__ w,
            float* __restrict__ tmp) {
    __shared__ float gs[KS + 3];
    const int t = threadIdx.x;
    load_g(w, gs, t);

    const int lane = t & 31;
    const int id   = blockIdx.x * 8 + (t >> 5);   // 16384 strips
    const int r0   = (id >> 6) << 4;              // tile row 0..255
    const int q0   = (id & 63) << 6;              // col block 0..63 (64 wide)

    // columns touched: [q0-50, q0+113]
    if (q0 >= 64 && q0 < IMG - 64)
        hblur_strip<false>(x, tmp, gs, r0, q0, lane);
    else
        hblur_strip<true>(x, tmp, gs, r0, q0, lane);
}

__global__ void __launch_bounds__(256)
gauss_vpass(const float* __restrict__ tmp, const float* __restrict__ w,
            float* __restrict__ out) {
    __shared__ float gs[KS + 3];
    const int t = threadIdx.x;
    load_g(w, gs, t);

    const int lane = t & 31;
    const int id   = blockIdx.x * 8 + (t >> 5);   // 16384 strips
    const int q0   = (id >> 8) << 6;              // row block 0..63 (64 tall)
    const int c0   = (id & 255) << 4;             // tile col 0..255

    // rows touched: [q0-50, q0+113]
    if (q0 >= 64 && q0 < IMG - 64)
        vblur_strip<false>(tmp, out, gs, q0, c0, lane);
    else
        vblur_strip<true>(tmp, out, gs, q0, c0, lane);
}

extern "C" void kernel_launch(void* const* d_in, const int* in_sizes, int n_in,
                              void* d_out, int out_size, void* d_ws, size_t ws_size,
                              hipStream_t stream) {
    const float* x   = (const float*)d_in[0];   // 4096*4096 fp32
    const float* w   = (const float*)d_in[1];   // 101*101 fp32 (outer(g,g))
    float*       out = (float*)d_out;           // 4096*4096 fp32
    float*       tmp = (float*)d_ws;            // 64 MB intermediate

    const int strips = (IMG / 16) * (IMG / 64); // 16384 per pass
    const dim3 blk(256);                        // 8 waves, one strip each
    const dim3 grd(strips / 8);                 // 2048 blocks

    gauss_hpass<<<grd, blk, 0, stream>>>(x, w, tmp);
    gauss_vpass<<<grd, blk, 0, stream>>>(tmp, w, out);
}